// ButterflyTransform_59322088292571
// MI455X (gfx1250) — compile-verified
//
#include <hip/hip_runtime.h>
#include <cstdint>

// ---------------------------------------------------------------------------
// Butterfly transform (12 layers, DIM=4096, BATCH=8192, fp32)
// Strategy: compose each group of 4 layers into 16x16 fp32 matrices and apply
// them with V_WMMA_F32_16X16X4_F32.  3 rounds per row, fused in one kernel
// with the 16-row block tile resident in LDS (padded stride 4097 floats).
// Input/output rows moved with the Tensor Data Mover (tensor_load_to_lds /
// tensor_store_from_lds), synchronized with s_wait_tensorcnt + barriers.
// ---------------------------------------------------------------------------

typedef float v2f  __attribute__((ext_vector_type(2)));
typedef float v8f  __attribute__((ext_vector_type(8)));
typedef unsigned int u32x4 __attribute__((ext_vector_type(4)));
typedef int  i32x4 __attribute__((ext_vector_type(4)));
typedef int  i32x8 __attribute__((ext_vector_type(8)));

#define DIMN       4096
#define NROWS      16           // batch rows per block (WMMA M dimension)
#define PSTRIDE    4097         // padded LDS row stride in floats (odd => conflict-free)
#define NMAT       273          // 256 (round0) + 16 (round1) + 1 (round2)

// ---------------------------------------------------------------------------
// TDM: one 1-row 2D descriptor (4096 x 1 fp32 tile), per CDNA5 ISA ch.8.
// group0: count=1 | lds_addr | global_addr lo | global_addr hi + type=2
// group1: data_size=4B, tensor_dim0=4096, tensor_dim1=1, tile_dim0=4096,
//         tile_dim1=1, tensor_dim0_stride=4096.
// ---------------------------------------------------------------------------
__device__ __forceinline__ void tdm_row(unsigned lds_addr, unsigned long long ga, bool is_load) {
  u32x4 g0;
  g0[0] = 1u;                                               // count=1 (valid), user mode
  g0[1] = lds_addr;                                         // LDS byte address
  g0[2] = (unsigned)(ga & 0xffffffffull);                   // global_addr[31:0]
  g0[3] = ((unsigned)((ga >> 32) & 0x1ffffffull)) | (2u << 30); // ga[56:32] | type=2
  i32x8 g1;
  g1[0] = (int)(2u << 16);        // workgroup_mask=0, data_size=2 (4 bytes)
  g1[1] = (int)(4096u << 16);     // atomic_barrier_addr=0 | tensor_dim0[15:0]=4096
  g1[2] = (int)(1u << 16);        // tensor_dim0[31:16]=0 | tensor_dim1[15:0]=1
  g1[3] = (int)(4096u << 16);     // tensor_dim1[31:16]=0 | tile_dim0=4096
  g1[4] = 1;                      // tile_dim1=1 | tile_dim2=0
  g1[5] = 4096;                   // tensor_dim0_stride[31:0]
  g1[6] = 0;                      // stride0[47:32]=0 | tensor_dim1_stride[15:0]=0
  g1[7] = 0;                      // tensor_dim1_stride[47:16]=0
  i32x4 z4 = {};
#if __has_include(<hip/amd_detail/amd_gfx1250_TDM.h>)
  i32x8 z8 = {};
  if (is_load) __builtin_amdgcn_tensor_load_to_lds(g0, g1, z4, z4, z8, 0);
  else         __builtin_amdgcn_tensor_store_from_lds(g0, g1, z4, z4, z8, 0);
#else
  if (is_load) __builtin_amdgcn_tensor_load_to_lds(g0, g1, z4, z4, 0);
  else         __builtin_amdgcn_tensor_store_from_lds(g0, g1, z4, z4, 0);
#endif
}

// ---------------------------------------------------------------------------
// Kernel 1: build the 273 composed 16x16 matrices, stored directly in the
// wave32 B-operand order:  float2 per lane per K-chunk:
//   mats[mat*128 + kc*32 + lane] = { Mt[kc*4 + 2*(lane/16)    ][lane%16],
//                                    Mt[kc*4 + 2*(lane/16) + 1][lane%16] }
// where Mt[k][j] = M[j][k] and  out[j] = sum_k M[j][k] * in[k].
// Thread k of a 16-thread block computes column k of M (= row k of Mt) by
// pushing the basis vector e_k through the round's 4 butterfly layers.
// ---------------------------------------------------------------------------
__global__ void build_mats_kernel(const float* __restrict__ a_flat,
                                  const float* __restrict__ b_flat,
                                  float* __restrict__ mats) {
  const int mat = blockIdx.x;     // 0..272
  const int k   = threadIdx.x;    // 0..15
  int r, q;
  if (mat < 256)      { r = 0; q = mat; }
  else if (mat < 272) { r = 1; q = mat - 256; }
  else                { r = 2; q = 0; }

  float c[16];
#pragma unroll
  for (int j = 0; j < 16; ++j) c[j] = (j == k) ? 1.0f : 0.0f;

  // coefficient offsets of layer l in a_flat/b_flat
  const int offs[12] = {0, 2048, 3072, 3584, 3840, 3968, 4032, 4064, 4080, 4088, 4092, 4094};

#pragma unroll
  for (int t = 0; t < 4; ++t) {
    const int l = 4 * r + t;
    const int s = 1 << t;
#pragma unroll
    for (int j = 0; j < 16; ++j) {
      if ((j & s) == 0) {
        const int j2  = j + s;
        const int idx = offs[l] + (j >> (t + 1)) + q * (16 >> (t + 1));
        const float a = a_flat[idx];
        const float b = b_flat[idx];
        const float x0 = c[j], x1 = c[j2];
        c[j]  =  a * x0 + b * x1;
        c[j2] = -b * x0 + a * x1;
      }
    }
  }

  // scatter row k of Mt into B-operand order
  const int kc = k >> 2, kk = k & 3;
  const int comp = kk & 1, lh = kk >> 1;
#pragma unroll
  for (int j = 0; j < 16; ++j) {
    const int lane = lh * 16 + j;
    mats[mat * 256 + kc * 64 + lane * 2 + comp] = c[j];
  }
}

// ---------------------------------------------------------------------------
// B-operand chunk load: one b64 per lane per chunk.
// ---------------------------------------------------------------------------
__device__ __forceinline__ void load_B(const float2* __restrict__ matsB, int mat,
                                       int lane, v2f Bv[4]) {
  const float2* p = matsB + mat * 128 + lane;
#pragma unroll
  for (int kc = 0; kc < 4; ++kc) {
    const float2 t = p[kc * 32];
    Bv[kc].x = t.x;
    Bv[kc].y = t.y;
  }
}

// ---------------------------------------------------------------------------
// One 16(batch) x 16(element-group) tile:  D[b, j] = sum_k X[b, k] * Mt[k][j]
// elements of the group at e(j) = ebase + j*estride in each LDS row.
// A layout (16x4 f32): lane%16 = batch row; VGPR0/1 = K {0,1} (lanes 0-15)
// or K {2,3} (lanes 16-31) of the current 4-wide K chunk.
// ---------------------------------------------------------------------------
__device__ __forceinline__ void tile_op(float* smem, int lane, int ebase,
                                        int estride, const v2f Bv[4]) {
  const int brow = lane & 15;
  const int half = lane >> 4;
  const float* rowp = smem + brow * PSTRIDE + ebase;

  v8f acc = {0.f, 0.f, 0.f, 0.f, 0.f, 0.f, 0.f, 0.f};
#pragma unroll
  for (int kc = 0; kc < 4; ++kc) {
    const int k0 = kc * 4 + 2 * half;
    v2f A;
    A.x = rowp[k0 * estride];
    A.y = rowp[(k0 + 1) * estride];
    acc = __builtin_amdgcn_wmma_f32_16x16x4_f32(
        false, A, false, Bv[kc], (short)0, acc, false, false);
  }

  // D: lane%16 = N(=j), VGPR v -> M(=batch row) = v + 8*(lane/16)
  float* wbase = smem + (half * 8) * PSTRIDE + ebase + brow * estride;
#pragma unroll
  for (int v = 0; v < 8; ++v)
    wbase[v * PSTRIDE] = acc[v];
}

// ---------------------------------------------------------------------------
// Main kernel: 512 blocks x 256 threads (8 wave32).  Each block owns 16 rows.
// ---------------------------------------------------------------------------
__global__ __launch_bounds__(256) void butterfly_kernel(
    const float* __restrict__ x, const float2* __restrict__ matsB,
    float* __restrict__ out) {
  __shared__ float smem[NROWS * PSTRIDE];   // 262,208 bytes

  const int tid  = threadIdx.x;
  const int wave = tid >> 5;
  const int lane = tid & 31;
  const long long base = (long long)blockIdx.x * NROWS * DIMN;

  // ---- stage 16 rows into LDS via TDM --------------------------------------
  if (wave == 0) {
    for (int r = 0; r < NROWS; ++r) {
      const unsigned lds = (unsigned)(uintptr_t)(const void*)&smem[r * PSTRIDE];
      tdm_row(lds, (unsigned long long)(uintptr_t)(x + base + (long long)r * DIMN), true);
    }
    __builtin_amdgcn_s_wait_tensorcnt(0);
  }
  __syncthreads();

  // ---- round 0: layers 0-3, e(j) = 16*T + j, matrix = T --------------------
  {
    v2f Bv[4];
    for (int T = wave * 32; T < wave * 32 + 32; ++T) {
      load_B(matsB, T, lane, Bv);
      tile_op(smem, lane, T * 16, 1, Bv);
    }
  }
  __syncthreads();

  // ---- round 1: layers 4-7, e(j) = lo + 16*j + 256*hi, matrix = 256+hi ----
  for (int hh = 0; hh < 2; ++hh) {
    const int hi = wave * 2 + hh;
    v2f Bv[4];
    load_B(matsB, 256 + hi, lane, Bv);
    for (int lo = 0; lo < 16; ++lo)
      tile_op(smem, lane, lo + 256 * hi, 16, Bv);
  }
  __syncthreads();

  // ---- round 2: layers 8-11, e(j) = T + 256*j, matrix = 272 ---------------
  {
    v2f Bv[4];
    load_B(matsB, 272, lane, Bv);
    for (int T = wave * 32; T < wave * 32 + 32; ++T)
      tile_op(smem, lane, T, 256, Bv);
  }
  __syncthreads();

  // ---- write 16 rows back via TDM -----------------------------------------
  if (wave == 0) {
    for (int r = 0; r < NROWS; ++r) {
      const unsigned lds = (unsigned)(uintptr_t)(const void*)&smem[r * PSTRIDE];
      tdm_row(lds, (unsigned long long)(uintptr_t)(out + base + (long long)r * DIMN), false);
    }
    // S_ENDPGM performs an implicit wait-idle (covers TENSORcnt).
  }
}

// ---------------------------------------------------------------------------
extern "C" void kernel_launch(void* const* d_in, const int* in_sizes, int n_in,
                              void* d_out, int out_size, void* d_ws, size_t ws_size,
                              hipStream_t stream) {
  const float* x      = (const float*)d_in[0];
  const float* a_flat = (const float*)d_in[1];
  const float* b_flat = (const float*)d_in[2];
  float* out  = (float*)d_out;
  float* mats = (float*)d_ws;          // 273 * 256 floats = 279,552 bytes

  build_mats_kernel<<<NMAT, 16, 0, stream>>>(a_flat, b_flat, mats);
  butterfly_kernel<<<8192 / NROWS, 256, 0, stream>>>(x, (const float2*)mats, out);
}